// GaussianVideo_57870389346587
// MI455X (gfx1250) — compile-verified
//
#include <hip/hip_runtime.h>
#include <cstdint>
#include <cstddef>

#define GV_N      20000
#define GV_NPAD   20032              // padded to multiple of 64
#define GV_W      128
#define GV_H      128
#define GV_T      8
#define GV_P      (GV_W * GV_H * GV_T)   // 131072 voxels
#define GV_CH     64                 // gaussians per LDS staging buffer
#define GV_NCHUNK (GV_NPAD / GV_CH)  // 313

typedef __attribute__((ext_vector_type(2))) float v2f;
typedef __attribute__((ext_vector_type(8))) float v8f;

// ---------------------------------------------------------------------------
// Kernel: WMMA sigma GEMM + alpha + color accumulation.
// Block = 256 threads = 8 waves; each wave owns 32 voxels (two 16-wide tiles).
// Gaussian records double-buffered in LDS via async global->LDS copies.
//
// Record layout (16 floats = 64 B), produced by gv_prep below:
//   [0..9]  polynomial coeffs of sigma over monomials
//           (1, px, py, pt, px^2, py^2, pt^2, px*py, px*pt, py*pt)
//   [10..11] zero pad (K padded 10->12); [12..14] RGB; [15] opacity (0 = pad)
// ---------------------------------------------------------------------------
__global__ void __launch_bounds__(256, 2)
gv_main(const float* __restrict__ Q, float* __restrict__ out) {
  __shared__ float lds[2 * GV_CH * 16];   // 2 x 4 KB staging buffers

  const int lane = threadIdx.x & 31;
  const int wave = threadIdx.x >> 5;
  const int n    = lane & 15;             // column (voxel) within a tile
  const int kh   = (lane >> 4) << 1;      // K sub-offset: 0 (lanes 0-15) / 2 (16-31)
  const int pbase = blockIdx.x * 256 + wave * 32;

  // Build B operands (monomial matrix, K=12 padded, N=16) for both tiles.
  v2f bm[2][3];
  int pix_x[2], pix_y[2], pix_t[2];
#pragma unroll
  for (int t2 = 0; t2 < 2; ++t2) {
    int p  = pbase + t2 * 16 + n;         // flat voxel index: p = t*H*W + y*W + x
    int x  = p & (GV_W - 1);
    int y  = (p >> 7) & (GV_H - 1);
    int tt = p >> 14;
    pix_x[t2] = x; pix_y[t2] = y; pix_t[t2] = tt;
    float px = x + 0.5f, py = y + 0.5f, pt = tt + 0.5f;
    float m[12] = {1.f, px, py, pt, px * px, py * py, pt * pt,
                   px * py, px * pt, py * pt, 0.f, 0.f};
#pragma unroll
    for (int j = 0; j < 3; ++j) {
      bm[t2][j][0] = m[4 * j + kh];
      bm[t2][j][1] = m[4 * j + kh + 1];
    }
  }

  float accx[2] = {0.f, 0.f}, accy[2] = {0.f, 0.f}, accz[2] = {0.f, 0.f};

  // Async stage of one 64-gaussian chunk (4096 B) into LDS buffer bufsel.
  // NOTE: the LDS byte offset is derived from the real &lds[...] pointer via
  // ptrtoint (low 32 bits of a generic LDS pointer == DS byte offset).  The
  // ptrtoint CAPTURES the shared array, so the "memory"-clobbering asm is a
  // potential writer of it and the staged-data loads cannot be folded away.
  auto stage = [&](int chunk, int bufsel) {
    const uint64_t src = (uint64_t)(uintptr_t)(Q + (size_t)chunk * GV_CH * 16);
    const uint32_t dstbase =
        (uint32_t)(uintptr_t)(lds + (size_t)bufsel * (GV_CH * 16));
#pragma unroll
    for (int i = 0; i < (GV_CH * 64) / (32 * 16); ++i) {   // 8 x b128 x 32 lanes
      uint32_t off = (uint32_t)(i * 32 + lane) * 16u;
      uint32_t la  = dstbase + off;
      uint64_t ga  = src + off;
      asm volatile("global_load_async_to_lds_b128 %0, %1, off"
                   :: "v"(la), "v"(ga) : "memory");
    }
  };

  if (wave == 0) {
    stage(0, 0);
    asm volatile("s_wait_asynccnt 0" ::: "memory");
  }
  __syncthreads();

  for (int c = 0; c < GV_NCHUNK; ++c) {
    const int cur = c & 1;
    if (wave == 0 && (c + 1) < GV_NCHUNK) stage(c + 1, cur ^ 1);  // prefetch
    const float* buf = lds + cur * (GV_CH * 16);

#pragma unroll 1
    for (int s = 0; s < GV_CH / 16; ++s) {
      const float* rec  = buf + s * 16 * 16;      // 16 gaussians x 16 floats
      const float* arow = rec + n * 16;           // this lane's gaussian row
      v2f a0, a1, a2;                             // A operand, K slices 0..3/4..7/8..11
      a0[0] = arow[kh];     a0[1] = arow[kh + 1];
      a1[0] = arow[4 + kh]; a1[1] = arow[4 + kh + 1];
      a2[0] = arow[8 + kh]; a2[1] = arow[8 + kh + 1];

      v8f sig[2];
#pragma unroll
      for (int t2 = 0; t2 < 2; ++t2) {
        v8f d = {0.f, 0.f, 0.f, 0.f, 0.f, 0.f, 0.f, 0.f};
        d = __builtin_amdgcn_wmma_f32_16x16x4_f32(false, a0, false, bm[t2][0],
                                                  (short)0, d, false, false);
        d = __builtin_amdgcn_wmma_f32_16x16x4_f32(false, a1, false, bm[t2][1],
                                                  (short)0, d, false, false);
        d = __builtin_amdgcn_wmma_f32_16x16x4_f32(false, a2, false, bm[t2][2],
                                                  (short)0, d, false, false);
        sig[t2] = d;
      }

      // alpha + color accumulation.  D layout: vgpr r, lanes 0-15 -> gaussian r,
      // lanes 16-31 -> gaussian r+8 (column = lane%16).
      const int mb = (lane >> 4) << 3;
#pragma unroll
      for (int r = 0; r < 8; ++r) {
        const float4 co = *(const float4*)(rec + (mb + r) * 16 + 12);  // RGB, op
#pragma unroll
        for (int t2 = 0; t2 < 2; ++t2) {
          float sg = sig[t2][r];
          float al = fminf(0.99f, co.w * __expf(-sg));
          al = (sg >= 0.f) ? al : 0.f;
          accx[t2] = fmaf(al, co.x, accx[t2]);
          accy[t2] = fmaf(al, co.y, accy[t2]);
          accz[t2] = fmaf(al, co.z, accz[t2]);
        }
      }
    }
    if (wave == 0) asm volatile("s_wait_asynccnt 0" ::: "memory");
    __syncthreads();
  }

  // Fold the two lane halves (same voxel, disjoint gaussian subsets), clip, store.
#pragma unroll
  for (int t2 = 0; t2 < 2; ++t2) {
    float rx = accx[t2] + __shfl_down(accx[t2], 16, 32);
    float ry = accy[t2] + __shfl_down(accy[t2], 16, 32);
    float rz = accz[t2] + __shfl_down(accz[t2], 16, 32);
    if (lane < 16) {
      size_t base = ((size_t)pix_y[t2] * GV_W + pix_x[t2]) * GV_T + pix_t[t2];
      const size_t CS = (size_t)GV_H * GV_W * GV_T;   // channel stride in (1,3,H,W,T)
      out[0 * CS + base] = fminf(fmaxf(rx, 0.f), 1.f);
      out[1 * CS + base] = fminf(fmaxf(ry, 0.f), 1.f);
      out[2 * CS + base] = fminf(fmaxf(rz, 0.f), 1.f);
    }
  }
}

// ---------------------------------------------------------------------------
// Per-gaussian preprocessing.  0.5*S*(tanh(x)+1) == S * sigmoid(2x)
//   -> one v_exp_f32 + one v_rcp per coordinate (no libm tanh).
// ---------------------------------------------------------------------------
__global__ void gv_prep(const float* __restrict__ xyz, const float* __restrict__ chol,
                        const float* __restrict__ feat, const float* __restrict__ opac,
                        float* __restrict__ Q) {
  int g = blockIdx.x * blockDim.x + threadIdx.x;
  if (g >= GV_NPAD) return;
  float q[16];
#pragma unroll
  for (int i = 0; i < 16; ++i) q[i] = 0.f;
  if (g < GV_N) {
    float cx = (float)GV_W / (1.f + __expf(-2.f * xyz[3 * g + 0]));
    float cy = (float)GV_H / (1.f + __expf(-2.f * xyz[3 * g + 1]));
    float ct = (float)GV_T / (1.f + __expf(-2.f * xyz[3 * g + 2]));
    float l0 = chol[6 * g + 0] + 0.5f;
    float l1 = chol[6 * g + 1];
    float l2 = chol[6 * g + 2] + 0.5f;
    float l3 = chol[6 * g + 3] + 0.5f;
    float l4 = chol[6 * g + 4];
    float l5 = chol[6 * g + 5] + 0.5f;
    // Sigma = L * L^T (symmetric 3x3)
    float S00 = l0 * l0, S01 = l0 * l1, S02 = l0 * l3;
    float S11 = l1 * l1 + l2 * l2;
    float S12 = l1 * l3 + l2 * l4;
    float S22 = l3 * l3 + l4 * l4 + l5 * l5;
    // Conic = Sigma^{-1} via adjugate
    float c00 = S11 * S22 - S12 * S12;
    float c01 = S02 * S12 - S01 * S22;
    float c02 = S01 * S12 - S02 * S11;
    float inv = 1.f / (S00 * c00 + S01 * c01 + S02 * c02);
    float A  = c00 * inv;
    float B  = c01 * inv;
    float Cc = (S00 * S22 - S02 * S02) * inv;
    float D  = c02 * inv;
    float E  = (S02 * S01 - S00 * S12) * inv;
    float F  = (S00 * S11 - S01 * S01) * inv;
    // sigma = 0.5(A dx^2 + C dy^2 + F dt^2) + B dxdy + D dxdt + E dydt,
    // dx = cx - px etc., expanded as a degree-2 polynomial in (px,py,pt).
    q[0] = 0.5f * (A * cx * cx + Cc * cy * cy + F * ct * ct)
         + B * cx * cy + D * cx * ct + E * cy * ct;
    q[1] = -(A * cx + B * cy + D * ct);
    q[2] = -(Cc * cy + B * cx + E * ct);
    q[3] = -(F * ct + D * cx + E * cy);
    q[4] = 0.5f * A;  q[5] = 0.5f * Cc;  q[6] = 0.5f * F;
    q[7] = B;  q[8] = D;  q[9] = E;
    q[12] = feat[3 * g + 0]; q[13] = feat[3 * g + 1]; q[14] = feat[3 * g + 2];
    q[15] = opac[g];
  }
  float4* dst = (float4*)(Q + 16 * (size_t)g);
#pragma unroll
  for (int i = 0; i < 4; ++i)
    dst[i] = make_float4(q[4 * i], q[4 * i + 1], q[4 * i + 2], q[4 * i + 3]);
}

// ---------------------------------------------------------------------------
extern "C" void kernel_launch(void* const* d_in, const int* in_sizes, int n_in,
                              void* d_out, int out_size, void* d_ws, size_t ws_size,
                              hipStream_t stream) {
  const float* xyz  = (const float*)d_in[0];
  const float* chol = (const float*)d_in[1];
  const float* feat = (const float*)d_in[2];
  const float* opac = (const float*)d_in[3];
  float* Q   = (float*)d_ws;                // 20032 * 64 B = 1.28 MB scratch
  float* out = (float*)d_out;

  gv_prep<<<(GV_NPAD + 255) / 256, 256, 0, stream>>>(xyz, chol, feat, opac, Q);
  gv_main<<<GV_P / 256, 256, 0, stream>>>(Q, out);

  (void)in_sizes; (void)n_in; (void)out_size; (void)ws_size;
}